// CustomLoss_47244640256592
// MI455X (gfx1250) — compile-verified
//
#include <hip/hip_runtime.h>

// ---------------------------------------------------------------------------
// Boundary-weighted BCE + Dice loss for (2,1,160,160,160) f32 logits / i32 target.
// Memory-bound reduction: ~65MB reads -> ~2.8us roofline @ 23.3 TB/s HBM.
// Neighbour re-reads of the target volume hit the 192MB L2 (dataset is 65MB).
// Final cross-lane reductions use V_WMMA_F32_16X16X4_F32 (A = per-lane partials,
// B = ones): exact f32 sums on the CDNA5 matrix pipe.
// ---------------------------------------------------------------------------

#define LAMBDA_BOUNDARY 1.0f
#define SMOOTH 1.0f

typedef __attribute__((ext_vector_type(2))) float v2f;
typedef __attribute__((ext_vector_type(8))) float v8f;

// Exact 32-lane f32 sum via WMMA 16x16x4:
//   A vgpr0 = x (lane m -> A[m][0], lane m+16 -> A[m][2]), A vgpr1 = 0,
//   B = ones  =>  D[m][n] = x[m] + x[m+16]  (replicated over n).
// D layout: lane<16 holds rowsum[r] in d[r] (r=0..7), lane>=16 holds rowsum[r+8].
// Sum the 8 D regs per lane, then xor-shuffle halves -> every lane has the total.
// Requires EXEC == all ones across the wave (call from a wave-uniform region).
__device__ __forceinline__ float wave32_sum_wmma(float x) {
  v2f a; a[0] = x;    a[1] = 0.0f;
  v2f b; b[0] = 1.0f; b[1] = 1.0f;
  v8f c = {};
  v8f d = __builtin_amdgcn_wmma_f32_16x16x4_f32(false, a, false, b,
                                                (short)0, c, false, false);
  float s = d[0] + d[1] + d[2] + d[3] + d[4] + d[5] + d[6] + d[7];
  s += __shfl_xor(s, 16, 32);
  return s;
}

// Grid geometry: each thread owns 4 consecutive W voxels.
constexpr int DIMB = 2, DIMD = 160, DIMH = 160, DIMW = 160;
constexpr int VPR  = DIMW / 4;                       // 40 vec4 per row
constexpr int NVEC = DIMB * DIMD * DIMH * VPR;       // 2,048,000 vec4 units
constexpr int BLK  = 256;
constexpr int NBLOCKS = NVEC / BLK;                  // 8000 partial float4s

__global__ void loss_partial_kernel(const float* __restrict__ logits,
                                    const int*   __restrict__ target,
                                    const float* __restrict__ area_table,
                                    float4*      __restrict__ partial) {
  __shared__ float  sArea[256];
  __shared__ float4 sred[BLK];
  const int t = threadIdx.x;
  sArea[t] = area_table[t & 255];
  __syncthreads();

  int idx = blockIdx.x * BLK + t;
  int w0 = (idx % VPR) * 4;
  int r  = idx / VPR;
  int h  = r % DIMH; r /= DIMH;
  int d  = r % DIMD;
  int b  = r / DIMD;

  const int rowS   = DIMW;
  const int planeS = DIMH * DIMW;
  const int volS   = DIMD * DIMH * DIMW;
  int base = b * volS + d * planeS + h * rowS + w0;

  // 16B-aligned vector loads (row stride 640B, w0 % 4 == 0).
  float4 lg  = *reinterpret_cast<const float4*>(logits + base);
  int4   t00 = *reinterpret_cast<const int4*>(target + base);

  // Neighbour rows (clamped at the volume edge; weight is zeroed there anyway).
  int d1 = d < DIMD - 1 ? d + 1 : d;
  int h1 = h < DIMH - 1 ? h + 1 : h;
  int base01 = b * volS + d  * planeS + h1 * rowS + w0;
  int base10 = b * volS + d1 * planeS + h  * rowS + w0;
  int base11 = b * volS + d1 * planeS + h1 * rowS + w0;
  int4 t01 = *reinterpret_cast<const int4*>(target + base01);
  int4 t10 = *reinterpret_cast<const int4*>(target + base10);
  int4 t11 = *reinterpret_cast<const int4*>(target + base11);

  // 5th window element (w0+4); not needed for the last group (w0 == 156).
  bool eok = (w0 + 4) < DIMW;
  int e00 = eok ? target[base   + 4] : 0;
  int e01 = eok ? target[base01 + 4] : 0;
  int e10 = eok ? target[base10 + 4] : 0;
  int e11 = eok ? target[base11 + 4] : 0;

  int a00[5] = {t00.x, t00.y, t00.z, t00.w, e00};
  int a01[5] = {t01.x, t01.y, t01.z, t01.w, e01};
  int a10[5] = {t10.x, t10.y, t10.z, t10.w, e10};
  int a11[5] = {t11.x, t11.y, t11.z, t11.w, e11};
  float lgs[4] = {lg.x, lg.y, lg.z, lg.w};

  bool dhok = (d < DIMD - 1) && (h < DIMH - 1);
  float accB = 0.f, accP = 0.f, accI = 0.f, accT = 0.f;
#pragma unroll
  for (int e = 0; e < 4; ++e) {
    float l  = lgs[e];
    float tf = (float)a00[e];
    // bce = max(l,0) - l*t + log1p(exp(-|l|))
    float bce = fmaxf(l, 0.f) - l * tf + log1pf(expf(-fabsf(l)));
    float p   = 1.f / (1.f + expf(-l));
    float wgt = 0.f;
    if (dhok && (w0 + e) < DIMW - 1) {
      int code = (a00[e] << 7) | (a00[e + 1] << 6) |
                 (a01[e] << 5) | (a01[e + 1] << 4) |
                 (a10[e] << 3) | (a10[e + 1] << 2) |
                 (a11[e] << 1) | (a11[e + 1]);
      wgt = sArea[code];
    }
    accB += (1.f + LAMBDA_BOUNDARY * wgt) * bce;
    accP += p;
    accI += p * tf;
    accT += tf;
  }

  sred[t] = make_float4(accB, accP, accI, accT);
  __syncthreads();
  for (int off = BLK / 2; off >= 1; off >>= 1) {
    if (t < off) {
      float4 o = sred[t + off];
      float4 m = sred[t];
      m.x += o.x; m.y += o.y; m.z += o.z; m.w += o.w;
      sred[t] = m;
    }
    __syncthreads();
  }
  if (t == 0) partial[blockIdx.x] = sred[0];
}

__global__ void loss_final_kernel(const float4* __restrict__ partial,
                                  int nblocks,
                                  float* __restrict__ out) {
  __shared__ float4 sred[BLK];
  const int t = threadIdx.x;
  float accB = 0.f, accP = 0.f, accI = 0.f, accT = 0.f;
  for (int i = t; i < nblocks; i += BLK) {
    float4 v = partial[i];
    accB += v.x; accP += v.y; accI += v.z; accT += v.w;
  }
  sred[t] = make_float4(accB, accP, accI, accT);
  __syncthreads();
  for (int off = BLK / 2; off >= 32; off >>= 1) {
    if (t < off) {
      float4 o = sred[t + off];
      float4 m = sred[t];
      m.x += o.x; m.y += o.y; m.z += o.z; m.w += o.w;
      sred[t] = m;
    }
    __syncthreads();
  }
  // Wave 0 only: uniform branch => EXEC all ones inside, legal for WMMA.
  if (t < 32) {
    float4 v = sred[t];
    float sB = wave32_sum_wmma(v.x);
    float sP = wave32_sum_wmma(v.y);
    float sI = wave32_sum_wmma(v.z);
    float sT = wave32_sum_wmma(v.w);
    if (t == 0) {
      const float N = (float)((long long)DIMB * DIMD * DIMH * DIMW);
      float wbce = sB / N;
      float dice = 1.0f - (2.0f * sI + SMOOTH) / (sP + sT + SMOOTH);
      out[0] = wbce + dice;
    }
  }
}

extern "C" void kernel_launch(void* const* d_in, const int* in_sizes, int n_in,
                              void* d_out, int out_size, void* d_ws, size_t ws_size,
                              hipStream_t stream) {
  const float* logits = (const float*)d_in[0];
  const int*   target = (const int*)d_in[1];
  const float* area   = (const float*)d_in[2];
  float4* partial = (float4*)d_ws;   // 8000 * 16B = 128KB scratch, fully rewritten

  loss_partial_kernel<<<NBLOCKS, BLK, 0, stream>>>(logits, target, area, partial);
  loss_final_kernel<<<1, BLK, 0, stream>>>(partial, NBLOCKS, (float*)d_out);
}